// Propagation_76106820485556
// MI455X (gfx1250) — compile-verified
//
#include <hip/hip_runtime.h>

// Problem constants (from the reference)
constexpr int Bn   = 4;
constexpr int Dn   = 8;
constexpr int Hn   = 512;
constexpr int Wn   = 640;
constexpr int NNb  = 9;            // neighbors
constexpr int Cout = Dn + NNb;     // 17 sorted channels out
constexpr int PLANE = Hn * Wn;     // 327680

typedef float vf2  __attribute__((ext_vector_type(2)));
typedef float vf4  __attribute__((ext_vector_type(4)));
// 4-byte-aligned pair for the (possibly odd-offset) bilinear gathers
typedef float vf2u __attribute__((ext_vector_type(2), aligned(4)));

__device__ __forceinline__ void cswap(float &a, float &b) {
  // v_min_num_f32 / v_max_num_f32, branch-free; the two independent
  // per-pixel sorts co-issue as v_dual_* pairs.
  float lo = fminf(a, b);
  float hi = fmaxf(a, b);
  a = lo;
  b = hi;
}

// Batcher odd-even mergesort, fully unrolled (constant indices -> registers).
template <int N>
__device__ __forceinline__ void sort_network(float (&v)[N]) {
#pragma unroll
  for (int p = 1; p < N; p <<= 1) {
#pragma unroll
    for (int k = p; k >= 1; k >>= 1) {
#pragma unroll
      for (int j = k % p; j + k < N; j += 2 * k) {
#pragma unroll
        for (int i = 0; i < k; ++i) {
          if (i + j + k < N) {
            if ((i + j) / (2 * p) == (i + j + k) / (2 * p)) {
              cswap(v[i + j], v[i + j + k]);
            }
          }
        }
      }
    }
  }
}

// Bilinear sample, padding_mode='border', align_corners=False.
// Border trick: x1 differs from x0+1 (resp. y1 from y0+1) only when the
// corresponding weight is exactly 0, and one element / one row past the
// center slice is still inside the depth buffer (channel 5 follows channel
// 4) with finite data -> load [v00,v01] and [v10,v11] as raw b64 pairs.
__device__ __forceinline__ float bilin(const float* __restrict__ c,
                                       float gx, float gy) {
  float ix = fminf(fmaxf(fmaf(gx, 0.5f * Wn, 0.5f * Wn - 0.5f), 0.0f),
                   (float)(Wn - 1));
  float iy = fminf(fmaxf(fmaf(gy, 0.5f * Hn, 0.5f * Hn - 0.5f), 0.0f),
                   (float)(Hn - 1));
  float x0f = floorf(ix), y0f = floorf(iy);
  float wx = ix - x0f, wy = iy - y0f;
  // flat offset y0*W + x0 computed exactly in f32 (max 327679 < 2^24)
  int addr = (int)fmaf(y0f, (float)Wn, x0f);
  vf2u r0 = *(const vf2u*)(c + addr);         // [v00, v01]
  vf2u r1 = *(const vf2u*)(c + addr + Wn);    // [v10, v11]
  float top = fmaf(wx, r0.y - r0.x, r0.x);
  float bot = fmaf(wx, r1.y - r1.x, r1.x);
  return fmaf(wy, bot - top, top);
}

// Block = 320 threads (10 wave32), one (b,h) row per block, 2 pixels/thread:
//   w = 2*threadIdx.x, 2*threadIdx.x + 1   (320*2 == 640 == W, exact)
// grid = (H, B). No integer division anywhere.
__global__ __launch_bounds__(320, 1) void prop_sort_kernel(
    const float* __restrict__ depth,   // [B, D, H, W]
    const float* __restrict__ grid,    // [B, NN*H, W, 2]
    float* __restrict__ out) {         // [B, 17, H, W]
  const int w0 = (int)threadIdx.x * 2;
  const int h  = (int)blockIdx.x;
  const int b  = (int)blockIdx.y;
  const int rowbase = h * Wn + w0;     // even -> 8B-aligned accesses

  float va[Cout];   // pixel w0
  float vb[Cout];   // pixel w0+1

  // ---- 8 depth channels: one coalesced b64 load per channel ----
  const vf2* dp = (const vf2*)(depth + b * (Dn * PLANE) + rowbase);
#pragma unroll
  for (int d = 0; d < Dn; ++d) {
    vf2 t = dp[d * (PLANE / 2)];
    va[d] = t.x;
    vb[d] = t.y;
  }

  const float* center = depth + (b * Dn + Dn / 2) * PLANE;

  // ---- 9 neighbors: one b128 NT grid load covers both pixels ----
  // float offset ((b*NN*H + h)*W + w0)*2 is a multiple of 4 -> 16B aligned.
  const vf4* gp =
      (const vf4*)(grid + ((size_t)(b * NNb * Hn + h) * Wn + w0) * 2);
#pragma unroll
  for (int n = 0; n < NNb; ++n) {
    // stream-ahead hint -> global_prefetch_b8 (no counter traffic)
    __builtin_prefetch(gp + n * (PLANE / 2) + 1024, 0, 0);
    // grid is single-use: non-temporal keeps the center slice cached
    vf4 g = __builtin_nontemporal_load(gp + n * (PLANE / 2));
    va[Dn + n] = bilin(center, g.x, g.y);
    vb[Dn + n] = bilin(center, g.z, g.w);
  }

  // ---- two independent register-resident sorts (dual-issue fodder) ----
  sort_network(va);
  sort_network(vb);

  // ---- 17 coalesced b64 non-temporal stores ----
  float* ob = out + b * (Cout * PLANE) + rowbase;
#pragma unroll
  for (int c = 0; c < Cout; ++c) {
    vf2 t;
    t.x = va[c];
    t.y = vb[c];
    __builtin_nontemporal_store(t, (vf2*)(ob + c * PLANE));
  }
}

extern "C" void kernel_launch(void* const* d_in, const int* in_sizes, int n_in,
                              void* d_out, int out_size, void* d_ws, size_t ws_size,
                              hipStream_t stream) {
  const float* depth = (const float*)d_in[0];  // B*D*H*W f32
  const float* grid  = (const float*)d_in[1];  // B*NN*H*W*2 f32
  float* out = (float*)d_out;                  // B*17*H*W f32

  dim3 block(320, 1, 1);        // 10 wave32 waves; 320*2 px = one W row
  dim3 grd(Hn, Bn, 1);          // 512 * 4 = 2048 blocks
  prop_sort_kernel<<<grd, block, 0, stream>>>(depth, grid, out);
}